// Cross_attention_2_65549790871666
// MI455X (gfx1250) — compile-verified
//
#include <hip/hip_runtime.h>
#include <hip/hip_bf16.h>

// ---------------------------------------------------------------------------
// Cross_attention_2 for MI455X (gfx1250, wave32, WMMA f32 16x16x4,
// async global->LDS staging in the attention GEMM).
//
// Shapes: C=32, D=36, HW=2304, P=9, P2=81, L=1024.
// Padded patch stride KP=84 (zero cols 81..83) -> 21 exact K-steps of 4.
// res_trans folded:  (V @ W1^T) @ W2^T == V @ (W2@W1)^T   (bias-free linears)
// ---------------------------------------------------------------------------

typedef float v2f __attribute__((ext_vector_type(2)));
typedef float v8f __attribute__((ext_vector_type(8)));

#define C_CH   32
#define NVOX   82944      // 36*2304
#define LPAT   1024
#define KP     84         // padded patch length (81 -> 84)
#define VROWS  32768      // C*L rows of V / P
#define VELEMS (VROWS * KP)   // 2,752,512 floats per V buffer

// D = A(16x4) * B(4x16) + C  -- exact fp32 matrix op on CDNA5
__device__ __forceinline__ v8f wmma_f32_k4(v2f a, v2f b, v8f c) {
    return __builtin_amdgcn_wmma_f32_16x16x4_f32(
        /*neg_a=*/false, a, /*neg_b=*/false, b,
        /*c_mod=*/(short)0, c, /*reuse_a=*/false, /*reuse_b=*/false);
}

// Load a 16x4 fragment (works for both A and "B = rowmajor-matrix^T"):
// lane L holds row (row0 + L&15), k = 4*kk + 2*(L>>4) + {0,1}
// (float2, 8B aligned: ld even, k even).
__device__ __forceinline__ v2f frag_ld(const float* __restrict__ p,
                                       int row0, int ld, int kk, int lane) {
    int r = row0 + (lane & 15);
    int k = (kk << 2) + ((lane >> 4) << 1);
    return *(const v2f*)(p + (size_t)r * ld + k);
}

// Same fragment pattern out of an LDS tile with row stride KP.
// Bank-conflict-free: bank = (r*84 + k) % 64; r*20 mod 64 injective over 16 rows.
__device__ __forceinline__ v2f frag_lds(const float* s, int row0, int kk, int lane) {
    int r = row0 + (lane & 15);
    int k = (kk << 2) + ((lane >> 4) << 1);
    return *(const v2f*)(s + r * KP + k);
}

// 16B global -> LDS asynchronous copy (ASYNCcnt-tracked, no VGPR round trip).
// LDS address = low 32 bits of the generic pointer (ISA flat-aperture rule).
__device__ __forceinline__ void async_g2l_b128(float* lds_dst, const float* gsrc) {
    unsigned int ldst = (unsigned int)(uintptr_t)lds_dst;
    asm volatile("global_load_async_to_lds_b128 %0, %1, off"
                 :: "v"(ldst), "v"(gsrc)
                 : "memory");
}

__device__ __forceinline__ void wait_asynccnt0() {
    asm volatile("s_wait_asynccnt 0x0" ::: "memory");
}

// ---------------------------------------------------------------------------
// Kernel 1: 1x1x1 conv (32x32 channel mix) fused with unfold into padded V.
// One thread per voxel, all 32 output channels. Also zeroes pad cols 81..83.
// ---------------------------------------------------------------------------
__global__ void conv_unfold_kernel(const float* __restrict__ x,
                                   const float* __restrict__ Wm,
                                   const float* __restrict__ bm,
                                   float* __restrict__ V) {
    __shared__ float Ws[C_CH * C_CH];
    __shared__ float bs[C_CH];
    int tid = threadIdx.x;
    for (int i = tid; i < C_CH * C_CH; i += 256) Ws[i] = Wm[i];
    if (tid < C_CH) bs[tid] = bm[tid];
    __syncthreads();

    int v = blockIdx.x * 256 + tid;        // grid = 324 blocks -> exactly NVOX
    float acc[C_CH];
#pragma unroll
    for (int c = 0; c < C_CH; ++c) acc[c] = 0.0f;
    for (int cp = 0; cp < C_CH; ++cp) {
        float xv = x[(size_t)cp * NVOX + v];
#pragma unroll
        for (int c = 0; c < C_CH; ++c)
            acc[c] = fmaf(Ws[c * C_CH + cp], xv, acc[c]);
    }

    // unfold: v -> (d, hw) -> patch l = pd*256+pw, intra k = kd*9+kw
    int d  = v / 2304,  hw = v - d * 2304;
    int pd = d / 9,     kd = d - pd * 9;
    int pw = hw / 9,    kw = hw - pw * 9;
    int l  = pd * 256 + pw;
    int k  = kd * 9 + kw;
#pragma unroll
    for (int c = 0; c < C_CH; ++c)
        V[(size_t)(c * LPAT + l) * KP + k] = acc[c] + bs[c];

    // zero the K padding (cols 81..83) for rows [0, VROWS)
    if (v < VROWS) {
        float* p = V + (size_t)v * KP;
        p[81] = 0.0f; p[82] = 0.0f; p[83] = 0.0f;
    }
}

// ---------------------------------------------------------------------------
// Kernel 2: Wc = W2 @ W1  (81x81), stored padded to 96x84 with zeros so WMMA
// B-fragments (rows up to 95, cols up to 83) read zeros out of range.
// ---------------------------------------------------------------------------
__global__ void combine_w_kernel(const float* __restrict__ W1,   // (162,81)
                                 const float* __restrict__ W2,   // (81,162)
                                 float* __restrict__ Wc) {       // (96,84)
    int id = blockIdx.x * 256 + threadIdx.x;
    if (id >= 96 * KP) return;
    int k2 = id / KP, k = id - k2 * KP;
    float s = 0.0f;
    if (k2 < 81 && k < 81) {
        for (int m = 0; m < 162; ++m)
            s = fmaf(W2[k2 * 162 + m], W1[m * 81 + k], s);
    }
    Wc[id] = s;
}

// ---------------------------------------------------------------------------
// Kernel 3: P = leaky_relu(V @ Wc^T, 0.2), in place on V.
// One wave per 16-row M-tile; 6 N-tiles cover n=0..95 (store-guard n<81);
// 21 K-steps cover padded K=84. Pad cols 81..83 of V stay zero.
// ---------------------------------------------------------------------------
__global__ void res_trans_kernel(float* __restrict__ V,
                                 const float* __restrict__ Wc) {
    int lane = threadIdx.x & 31;
    int wave = threadIdx.x >> 5;
    int row0 = (blockIdx.x * 8 + wave) * 16;   // grid = 256 blocks, 8 waves ea.

    v8f acc[6];
#pragma unroll
    for (int n = 0; n < 6; ++n) acc[n] = (v8f){};

    for (int kk = 0; kk < 21; ++kk) {
        v2f a = frag_ld(V, row0, KP, kk, lane);
#pragma unroll
        for (int n = 0; n < 6; ++n) {
            v2f b = frag_ld(Wc, n * 16, KP, kk, lane);   // B[k][n] = Wc[n][k]
            acc[n] = wmma_f32_k4(a, b, acc[n]);
        }
    }

    int nlo = lane & 15;
    int mhi = (lane >> 4) << 3;   // 0 or 8
#pragma unroll
    for (int n = 0; n < 6; ++n) {
        int col = n * 16 + nlo;
        if (col < 81) {
#pragma unroll
            for (int r = 0; r < 8; ++r) {
                float val = acc[n][r];
                val = val > 0.0f ? val : 0.2f * val;
                V[(size_t)(row0 + r + mhi) * KP + col] = val;
            }
        }
    }
}

// ---------------------------------------------------------------------------
// Kernel 4: att[c] = PX[c] @ PY[c]^T / 81   (per channel 1024x1024, K=84 pad).
// WG = 256 thr = 8 waves (4 x 2); wave tile 32x32 (2x2 WMMA accums);
// WG tile 128 x 64; grid (8, 16, 32).
// PX/PY tiles staged once per WG into LDS via async global->LDS b128 copies
// (64.5 KB LDS), fragments then served conflict-free by ds_load_b64.
// ---------------------------------------------------------------------------
__global__ void att_kernel(const float* __restrict__ PX,
                           const float* __restrict__ PY,
                           float* __restrict__ out) {
    __shared__ __align__(16) float sx[128 * KP];   // 43,008 B
    __shared__ __align__(16) float sy[64 * KP];    // 21,504 B

    int tid  = threadIdx.x;
    int lane = tid & 31;
    int wave = tid >> 5;
    int wl1  = wave >> 1;          // 0..3
    int wl2  = wave & 1;           // 0..1
    int c    = blockIdx.z;
    int l1blk = blockIdx.x * 128;
    int l2blk = blockIdx.y * 64;

    const float* px = PX + (size_t)c * LPAT * KP + (size_t)l1blk * KP;
    const float* py = PY + (size_t)c * LPAT * KP + (size_t)l2blk * KP;

    // Stage tiles: 128*84 + 64*84 floats = 2688 + 1344 = 4032 16B packets.
    const int NX4 = 128 * KP / 4;          // 2688
    const int NTOT = NX4 + 64 * KP / 4;    // 4032
    for (int i = tid; i < NTOT; i += 256) {
        if (i < NX4) async_g2l_b128(sx + i * 4, px + i * 4);
        else         async_g2l_b128(sy + (i - NX4) * 4, py + (i - NX4) * 4);
    }
    wait_asynccnt0();
    __syncthreads();

    int r1 = wl1 * 32;   // local l1 base in sx
    int r2 = wl2 * 32;   // local l2 base in sy

    v8f acc00 = (v8f){}, acc01 = (v8f){}, acc10 = (v8f){}, acc11 = (v8f){};

    for (int kk = 0; kk < 21; ++kk) {
        v2f a0 = frag_lds(sx, r1,      kk, lane);
        v2f a1 = frag_lds(sx, r1 + 16, kk, lane);
        v2f b0 = frag_lds(sy, r2,      kk, lane);   // B[k][n] = PY[n][k]
        v2f b1 = frag_lds(sy, r2 + 16, kk, lane);
        acc00 = wmma_f32_k4(a0, b0, acc00);
        acc01 = wmma_f32_k4(a0, b1, acc01);
        acc10 = wmma_f32_k4(a1, b0, acc10);
        acc11 = wmma_f32_k4(a1, b1, acc11);
    }

    const float s = 1.0f / 81.0f;
    int nlo = lane & 15;
    int mhi = (lane >> 4) << 3;
    int l1_0 = l1blk + r1;
    int l2_0 = l2blk + r2;
    size_t base = (size_t)c << 20;   // c * 1024 * 1024

#pragma unroll
    for (int r = 0; r < 8; ++r) {
        int m0 = l1_0 + r + mhi;
        out[base + (size_t)m0        * LPAT + (l2_0 + nlo)]      = acc00[r] * s;
        out[base + (size_t)m0        * LPAT + (l2_0 + 16 + nlo)] = acc01[r] * s;
        out[base + (size_t)(m0 + 16) * LPAT + (l2_0 + nlo)]      = acc10[r] * s;
        out[base + (size_t)(m0 + 16) * LPAT + (l2_0 + 16 + nlo)] = acc11[r] * s;
    }
}

// ---------------------------------------------------------------------------
// Launch: inputs = { x, y, W_img, b_img, W_fea, b_fea, W1, W2 }
// ws: Vx (2,752,512 f) | Vy (2,752,512 f) | Wc (8,064 f)  ~= 22 MB
// ---------------------------------------------------------------------------
extern "C" void kernel_launch(void* const* d_in, const int* in_sizes, int n_in,
                              void* d_out, int out_size, void* d_ws, size_t ws_size,
                              hipStream_t stream) {
    const float* x     = (const float*)d_in[0];
    const float* y     = (const float*)d_in[1];
    const float* W_img = (const float*)d_in[2];
    const float* b_img = (const float*)d_in[3];
    const float* W_fea = (const float*)d_in[4];
    const float* b_fea = (const float*)d_in[5];
    const float* W1    = (const float*)d_in[6];
    const float* W2    = (const float*)d_in[7];
    float* out = (float*)d_out;

    float* ws = (float*)d_ws;
    float* Vx = ws;
    float* Vy = ws + (size_t)VELEMS;
    float* Wc = ws + (size_t)2 * VELEMS;

    conv_unfold_kernel<<<NVOX / 256, 256, 0, stream>>>(x, W_img, b_img, Vx);
    conv_unfold_kernel<<<NVOX / 256, 256, 0, stream>>>(y, W_fea, b_fea, Vy);
    combine_w_kernel<<<(96 * KP + 255) / 256, 256, 0, stream>>>(W1, W2, Wc);
    res_trans_kernel<<<VROWS / 128, 256, 0, stream>>>(Vx, Wc);   // 256 blocks
    res_trans_kernel<<<VROWS / 128, 256, 0, stream>>>(Vy, Wc);
    att_kernel<<<dim3(8, 16, 32), 256, 0, stream>>>(Vx, Vy, out);
}